// TrueRingDilatedAttention_69707319214698
// MI455X (gfx1250) — compile-verified
//
#include <hip/hip_runtime.h>

typedef __attribute__((ext_vector_type(2))) float v2f;
typedef __attribute__((ext_vector_type(4))) float v4f;
typedef __attribute__((ext_vector_type(8))) float v8f;

namespace {
constexpr int kB  = 2;
constexpr int kN  = 8192;
constexpr int kH  = 24;
constexpr int kD  = 128;
constexpr int kG  = 3;
constexpr int kHG = 8;                       // heads per group
constexpr int kWaves = 2;                    // waves per block; 1 problem-pair per wave
constexpr int kPairs = kB * (kN / 2) * kG;   // 24576 pairs of (b,n,g) problems
constexpr int kQKStride = 132;               // padded LDS row stride (floats): 528B rows -> 16B aligned, no bank conflicts
constexpr int kPStride  = 18;                // padded P-scratch stride: 8B-aligned b64 reads, conflict-free
constexpr float kScale = 0.088388347648318447f;  // 1/sqrt(128)
constexpr float kInvG  = 1.0f / 3.0f;
}

// 32-bit LDS byte offset of a generic pointer that targets __shared__ memory.
__device__ __forceinline__ unsigned lds_offset(const void* p) {
  return (unsigned)(unsigned long long)(__attribute__((address_space(3))) const char*)p;
}

// CDNA5 async copy: 16B global -> LDS per lane, tracked by ASYNCcnt.
// No VGPR staging, no ds_store; frees ~32 VGPRs/lane vs load+store staging.
__device__ __forceinline__ void async_load_b128(unsigned lds_byte_off, const void* gaddr) {
  asm volatile("global_load_async_to_lds_b128 %0, %1, off"
               :
               : "v"(lds_byte_off), "v"((unsigned long long)gaddr)
               : "memory");
}

__device__ __forceinline__ void wait_asynccnt0() {
  asm volatile("s_wait_asynccnt 0x0" ::: "memory");
}

__global__ __launch_bounds__(kWaves * 32)
void dilated_group_attn_wmma(const float* __restrict__ q,
                             const float* __restrict__ k,
                             const float* __restrict__ v,
                             float* __restrict__ out) {
  __shared__ __align__(16) float sQ[kWaves][16 * kQKStride];
  __shared__ __align__(16) float sK[kWaves][16 * kQKStride];
  __shared__ __align__(16) float sP[kWaves][16 * kPStride];

  const int lane = threadIdx.x & 31;
  const int wave = threadIdx.x >> 5;

  // pair = ((b * N/2 + n2) * G + g); wave handles positions n0 and n0+1 of group g
  const int pair = blockIdx.x * kWaves + wave;
  const int g  = pair % kG;
  const int t  = pair / kG;
  const int n2 = t % (kN / 2);
  const int b  = t / (kN / 2);
  const int n0 = n2 * 2;

  const long long base0 = ((((long long)b * kN + n0) * kH) + (long long)g * kHG) * kD;
  const long long base1 = base0 + (long long)kH * kD;   // position n0+1

  const float* v0 = v + base0;
  const float* v1 = v + base1;

  // V tiles are consumed last: prefetch them now (global_prefetch_b8).
  __builtin_prefetch(v0 + lane * 32, 0, 0);
  __builtin_prefetch(v1 + lane * 32, 0, 0);

  // ---- Stage Q/K tiles (two 8x128 problems stacked -> 16x128) into LDS ----
  // Async global->LDS b128 copies (ASYNCcnt path): coalesced 512B per
  // instruction across the wave, written straight into padded LDS rows.
  #pragma unroll
  for (int p = 0; p < 2; ++p) {
    const float* qp = q + (p ? base1 : base0);
    const float* kp = k + (p ? base1 : base0);
    #pragma unroll
    for (int i = 0; i < 8; ++i) {
      const int e   = (i * 32 + lane) * 4;   // tile-linear float index (16B aligned)
      const int col = e & (kD - 1);
      const int row = p * 8 + (e >> 7);      // == p*8 + i
      async_load_b128(lds_offset(&sQ[wave][row * kQKStride + col]), qp + e);
      async_load_b128(lds_offset(&sK[wave][row * kQKStride + col]), kp + e);
    }
  }
  wait_asynccnt0();   // staging complete before fragment reads (wave-private tiles)

  const int rowA = lane & 15;           // A/B fragment row for this lane
  const int rsel = (lane >> 4) << 1;    // K sub-offset: 0 (lanes 0-15) / 2 (lanes 16-31)
  const int ncol = lane & 15;           // C/D column held by this lane
  const int hi   = lane >> 4;           // which 8-row half of C/D this lane holds

  // ---- S[16x16] = Qtile * Ktile^T over d=128 via 32x v_wmma_f32_16x16x4_f32 ----
  v8f S = {};
  #pragma unroll
  for (int kk = 0; kk < kD; kk += 4) {
    v2f a  = *(const v2f*)&sQ[wave][rowA * kQKStride + kk + rsel];
    v2f bf = *(const v2f*)&sK[wave][rowA * kQKStride + kk + rsel];
    S = __builtin_amdgcn_wmma_f32_16x16x4_f32(false, a, false, bf, (short)0, S,
                                              false, false);
  }

  // ---- softmax across the 8-head axis (valid lane clusters: 0-7 and 24-31) ----
  // Off-diagonal (cross-problem) entries are zeroed -> P is block-diagonal.
  const bool valid = (lane < 8) || (lane >= 24);
  v8f P;
  #pragma unroll
  for (int j = 0; j < 8; ++j) {
    float s  = S[j] * kScale;
    float mx = s;
    mx = fmaxf(mx, __shfl_xor(mx, 1));
    mx = fmaxf(mx, __shfl_xor(mx, 2));
    mx = fmaxf(mx, __shfl_xor(mx, 4));
    float e = __expf(s - mx);
    float sum = e;
    sum += __shfl_xor(sum, 1);
    sum += __shfl_xor(sum, 2);
    sum += __shfl_xor(sum, 4);
    P[j] = valid ? (e / sum) * kInvG : 0.0f;
  }

  // C-layout -> A-layout transpose of P through wave-private LDS scratch
  #pragma unroll
  for (int j = 0; j < 8; ++j)
    sP[wave][(j + hi * 8) * kPStride + ncol] = P[j];

  v2f pa[4];
  #pragma unroll
  for (int kc = 0; kc < 4; ++kc)
    pa[kc] = *(const v2f*)&sP[wave][rowA * kPStride + kc * 4 + rsel];

  // ---- O[16x128] = P[16x16] * Vtile[16x128]; V B-fragments load coalesced
  // (16 consecutive floats per half-wave = 64B lines) straight from global ----
  float* op = out + (hi ? base1 : base0);
  #pragma unroll
  for (int nn = 0; nn < 8; ++nn) {
    v8f o = {};
    #pragma unroll
    for (int kc = 0; kc < 4; ++kc) {
      const int k0 = kc * 4 + rsel;
      const int k1 = k0 + 1;
      const float* vp0 = (k0 < 8) ? (v0 + k0 * kD) : (v1 + (k0 - 8) * kD);
      const float* vp1 = (k1 < 8) ? (v0 + k1 * kD) : (v1 + (k1 - 8) * kD);
      v2f bf;
      bf[0] = vp0[nn * 16 + ncol];
      bf[1] = vp1[nn * 16 + ncol];
      o = __builtin_amdgcn_wmma_f32_16x16x4_f32(false, pa[kc], false, bf, (short)0,
                                                o, false, false);
    }
    #pragma unroll
    for (int j = 0; j < 8; ++j)
      op[j * kD + nn * 16 + ncol] = o[j];   // 64B-contiguous per half-wave
  }
}

extern "C" void kernel_launch(void* const* d_in, const int* in_sizes, int n_in,
                              void* d_out, int out_size, void* d_ws, size_t ws_size,
                              hipStream_t stream) {
  const float* q = (const float*)d_in[0];
  const float* k = (const float*)d_in[1];
  const float* v = (const float*)d_in[2];
  float* out = (float*)d_out;
  dim3 grid(kPairs / kWaves);   // 12288 blocks
  dim3 block(kWaves * 32);      // 64 threads = 2 waves
  dilated_group_attn_wmma<<<grid, block, 0, stream>>>(q, k, v, out);
}